// ChannelAttention_54030688583941
// MI455X (gfx1250) — compile-verified
//
#include <hip/hip_runtime.h>

// Fused ChannelAttention for MI455X (gfx1250, wave32).
// qkv GEMM + per-token channel attention + proj GEMM in ONE kernel.
// Both weight matrices live in LDS (320KB/WGP) in WMMA-B-fragment order;
// HBM traffic = read x (32MB) + write out (32MB) ~ 2.7us floor @ 23.3TB/s.

#define DIM      128
#define NHEAD    8
#define NTOK     (4 * 16384)      // B*L = 65536 tokens
#define NROWT    (NTOK / 16)      // 4096 row tiles of 16 tokens
#define NT_QKV   24               // 384/16 N-tiles of qkv GEMM
#define NT_PROJ  8                // 128/16 N-tiles of proj GEMM

#define WAVES_PER_BLOCK 4
#define BLOCK   (WAVES_PER_BLOCK * 32)
#define GRID    512

typedef float v2f __attribute__((ext_vector_type(2)));
typedef float v8f __attribute__((ext_vector_type(8)));

// ---- LDS layout (float offsets) ----
#define OFF_WQKV  0
#define SZ_WQKV   (128 * 384)                       // 49152 floats, swizzled
#define OFF_WPROJ (OFF_WQKV + SZ_WQKV)              // 49152
#define SZ_WPROJ  (128 * 128)                       // 16384 floats, swizzled
#define OFF_ABUF  (OFF_WPROJ + SZ_WPROJ)            // 65536
#define SZ_ABUF_W (16 * DIM)                        // 2048 floats per wave
#define OFF_KV    (OFF_ABUF + WAVES_PER_BLOCK * SZ_ABUF_W)  // 73728
#define SZ_KV_W   (16 * 16 * 2)                     // 512 floats per wave
#define LDS_FLOATS (OFF_KV + WAVES_PER_BLOCK * SZ_KV_W)     // 75776 -> 303104 B

__launch_bounds__(BLOCK)
__global__ void fused_channel_attn(const float* __restrict__ x,
                                   const float* __restrict__ w_qkv,
                                   const float* __restrict__ b_qkv,
                                   const float* __restrict__ w_proj,
                                   const float* __restrict__ b_proj,
                                   float* __restrict__ out)
{
    __shared__ float lds[LDS_FLOATS];

    const int tid  = threadIdx.x;
    const int wave = tid >> 5;
    const int lane = tid & 31;
    const int l16  = lane & 15;
    const int half = lane >> 4;

    // ---- Phase 0: stage weights into LDS, pre-swizzled to the f32 WMMA
    // B-fragment layout (16x16x4): fragment (kt, nt) occupies 64 floats,
    // lane n (0..15) holds rows 4kt+0,4kt+1 at col nt*16+n; lane 16+n holds
    // rows 4kt+2,4kt+3. One ds_load_b64 per B fragment at compute time.
    for (int idx = tid; idx < 128 * 384; idx += BLOCK) {
        int k = idx / 384, n = idx % 384;
        int kt = k >> 2, kr = k & 3;
        int nt = n >> 4, nl = n & 15;
        lds[OFF_WQKV + (((kt * NT_QKV + nt) * 32 + (nl + 16 * (kr >> 1))) << 1)
                     + (kr & 1)] = w_qkv[idx];
    }
    for (int idx = tid; idx < 128 * 128; idx += BLOCK) {
        int k = idx / 128, n = idx % 128;
        int kt = k >> 2, kr = k & 3;
        int nt = n >> 4, nl = n & 15;
        lds[OFF_WPROJ + (((kt * NT_PROJ + nt) * 32 + (nl + 16 * (kr >> 1))) << 1)
                      + (kr & 1)] = w_proj[idx];
    }
    __syncthreads();

    // Per-lane bias preload: for N-tile nt, lane needs column nt*16 + l16.
    float bqk[NT_QKV];
#pragma unroll
    for (int nt = 0; nt < NT_QKV; ++nt) bqk[nt] = b_qkv[nt * 16 + l16];
    float bp[NT_PROJ];
#pragma unroll
    for (int nt = 0; nt < NT_PROJ; ++nt) bp[nt] = b_proj[nt * 16 + l16];

    float* abuf  = &lds[OFF_ABUF + wave * SZ_ABUF_W];   // x tile, then attn-out tile
    float* kvbuf = &lds[OFF_KV   + wave * SZ_KV_W];     // per-head K/V interleaved

    // softmax logits scale: C^-0.5, folded with log2(e) for v_exp_f32
    const float c1 = 0.08838834764831845f * 1.4426950408889634f;

    const int waveGlobal = blockIdx.x * WAVES_PER_BLOCK + wave;
    const int waveStride = GRID * WAVES_PER_BLOCK;

    for (int rt = waveGlobal; rt < NROWT; rt += waveStride) {
        const int row0 = rt * 16;

        // ---- stage x row tile (16x128) into abuf, coalesced b128 ----
        const float4* xg  = (const float4*)(x + (size_t)row0 * DIM);
        float4*       ab4 = (float4*)abuf;
#pragma unroll
        for (int i = 0; i < (16 * DIM / 4) / 32; ++i)   // 16 iters
            ab4[lane + i * 32] = xg[lane + i * 32];

        // ---- A fragments (16x4 f32 layout): lane l16 = row m,
        // half selects K pair {0,1} vs {2,3} within each 4-wide K tile ----
        v2f a[32];
#pragma unroll
        for (int kt = 0; kt < 32; ++kt)
            a[kt] = *(const v2f*)&abuf[l16 * DIM + kt * 4 + half * 2];

        // ---- per head: QKV tiles via WMMA, then channel attention ----
#pragma unroll 1
        for (int h = 0; h < NHEAD; ++h) {
            v8f cq, ck, cv;
#pragma unroll
            for (int r = 0; r < 8; ++r) {
                cq[r] = bqk[h];          // bias folded into accumulator init
                ck[r] = bqk[8 + h];
                cv[r] = bqk[16 + h];
            }
#pragma unroll
            for (int kt = 0; kt < 32; ++kt) {
                v2f bq = *(const v2f*)&lds[OFF_WQKV + (kt * NT_QKV + h     ) * 64 + lane * 2];
                v2f bk = *(const v2f*)&lds[OFF_WQKV + (kt * NT_QKV + h + 8 ) * 64 + lane * 2];
                v2f bv = *(const v2f*)&lds[OFF_WQKV + (kt * NT_QKV + h + 16) * 64 + lane * 2];
                cq = __builtin_amdgcn_wmma_f32_16x16x4_f32(false, a[kt], false, bq,
                                                           (short)0, cq, false, false);
                ck = __builtin_amdgcn_wmma_f32_16x16x4_f32(false, a[kt], false, bk,
                                                           (short)0, ck, false, false);
                cv = __builtin_amdgcn_wmma_f32_16x16x4_f32(false, a[kt], false, bv,
                                                           (short)0, cv, false, false);
            }

            // stash K,V interleaved: kvbuf[(t*16 + j)*2] = {k[t][j], v[t][j]}
            // C/D layout: VGPR r, lanes 0-15 -> token r, lanes 16-31 -> token r+8
#pragma unroll
            for (int r = 0; r < 8; ++r) {
                int t = r + 8 * half;
                v2f kv; kv.x = ck[r]; kv.y = cv[r];
                *(v2f*)&kvbuf[(t * 16 + l16) * 2] = kv;
            }

            // o[t][i] = sum_j exp(s*q[t][i]*k[t][j]) * v[t][j] / sum_j exp(...)
            // Output (t,i) indexing == C/D fragment indexing (t=M, i=N=l16).
#pragma unroll
            for (int r = 0; r < 8; ++r) {
                int t   = r + 8 * half;
                float q = cq[r];
                float num = 0.0f, den = 0.0f;
#pragma unroll
                for (int j = 0; j < 16; ++j) {
                    v2f kv  = *(const v2f*)&kvbuf[(t * 16 + j) * 2];
                    float e = __builtin_amdgcn_exp2f(q * kv.x * c1);
                    num += e * kv.y;
                    den += e;
                }
                // x tile data here is dead (A frags live in regs) -> reuse abuf
                abuf[t * DIM + h * 16 + l16] = num * __builtin_amdgcn_rcpf(den);
            }
        }

        // ---- proj GEMM: reload A frags from attention-output tile ----
#pragma unroll
        for (int kt = 0; kt < 32; ++kt)
            a[kt] = *(const v2f*)&abuf[l16 * DIM + kt * 4 + half * 2];

#pragma unroll 1
        for (int nt = 0; nt < NT_PROJ; ++nt) {
            v8f c;
#pragma unroll
            for (int r = 0; r < 8; ++r) c[r] = bp[nt];
#pragma unroll
            for (int kt = 0; kt < 32; ++kt) {
                v2f b = *(const v2f*)&lds[OFF_WPROJ + (kt * NT_PROJ + nt) * 64 + lane * 2];
                c = __builtin_amdgcn_wmma_f32_16x16x4_f32(false, a[kt], false, b,
                                                          (short)0, c, false, false);
            }
#pragma unroll
            for (int r = 0; r < 8; ++r)
                out[(size_t)(row0 + r + 8 * half) * DIM + nt * 16 + l16] = c[r];
        }
    }
}

extern "C" void kernel_launch(void* const* d_in, const int* in_sizes, int n_in,
                              void* d_out, int out_size, void* d_ws, size_t ws_size,
                              hipStream_t stream) {
    const float* x      = (const float*)d_in[0];
    const float* w_qkv  = (const float*)d_in[1];
    const float* b_qkv  = (const float*)d_in[2];
    const float* w_proj = (const float*)d_in[3];
    const float* b_proj = (const float*)d_in[4];
    float* out = (float*)d_out;
    (void)in_sizes; (void)n_in; (void)out_size; (void)d_ws; (void)ws_size;
    fused_channel_attn<<<dim3(GRID), dim3(BLOCK), 0, stream>>>(
        x, w_qkv, b_qkv, w_proj, b_proj, out);
}